// ChannelWiseMaxPoolWithCrossInfo_46875273068923
// MI455X (gfx1250) — compile-verified
//
#include <hip/hip_runtime.h>
#include <stdint.h>

// out[b][c2][c1][p] = x[b][c1][argmax2x2_flat_idx(b,c2,p)]
// B=8, C=128, H=W=64, k=2 -> Hp=Wp=32, P=1024. Output 512MB (write-bound,
// ~22.7us at 23.3 TB/s HBM). No matmul FLOPs -> no WMMA; CDNA5 paths used:
// GLOBAL_LOAD_ASYNC_TO_LDS_B128 (ASYNCcnt, per-wave double buffering, zero
// barriers since wave==pooled-row), DS gathers from LDS, non-temporal stores.

#define BB 8
#define CC 128
#define HH 64
#define WW 64
#define HP 32
#define WP 32
#define PP (HP * WP)   // 1024 pooled positions
#define PTILE 256      // pooled positions per block (= 8 pooled rows)
#define G 8            // c2 values per block (amortize LDS staging 8x)

__global__ __launch_bounds__(256)
void cwmp_gather(const float* __restrict__ x, float* __restrict__ out) {
    // Per buffer: 8 waves x (2 rows x 64 cols) floats = 4KB; double buffered.
    // Wave w stages and gathers ONLY rows [2w, 2w+2) of the tile -> no barriers.
    __shared__ __align__(16) float smem[2][8 * 128];

    const int t    = threadIdx.x;       // 0..255
    const int lane = t & 31;            // == pooled col pw (Wp == wave32 width)
    const int wv   = t >> 5;            // wave id == pooled row within tile
    const int b      = blockIdx.z;
    const int c2base = blockIdx.y * G;
    const int p0     = blockIdx.x * PTILE;
    const int ph     = (p0 + t) >> 5;                  // absolute pooled row
    const int rstart = blockIdx.x * ((PTILE / WP) * 2); // first original row of tile

    // ---- per-c2 argmax of the 2x2 window (first occurrence, row-major) ----
    int lidx[G];
#pragma unroll
    for (int j = 0; j < G; ++j) {
        const float* wsrc = x + (size_t)(b * CC + c2base + j) * (HH * WW)
                              + (size_t)(2 * ph) * WW + 2 * lane;
        float2 topr = *(const float2*)(wsrc);
        float2 botr = *(const float2*)(wsrc + WW);
        float best = topr.x; int bi = 0;
        if (topr.y > best) { best = topr.y; bi = 1; }
        if (botr.x > best) { best = botr.x; bi = 2; }
        if (botr.y > best) { best = botr.y; bi = 3; }
        // LDS offset inside this wave's 2x64 slice of the staged tile
        lidx[j] = wv * 128 + (bi >> 1) * WW + 2 * lane + (bi & 1);
    }

    // ---- async staging setup: lane l of wave w copies 16B of rows [2w,2w+2) ----
    const uint32_t lds0 = (uint32_t)(uintptr_t)(&smem[0][0]) + (uint32_t)t * 16u;
    const uint32_t voff = (uint32_t)t * 16u;  // 512*wv + 16*lane bytes into tile
    const float* srcb = x + (size_t)b * CC * (HH * WW) + (size_t)rstart * WW;

    float* orow[G];
#pragma unroll
    for (int j = 0; j < G; ++j)
        orow[j] = out + ((size_t)(b * CC + c2base + j) * CC) * PP + p0 + t;

    { // prologue: stage channel c1=0 into buffer 0
        uint64_t sa = (uint64_t)(uintptr_t)srcb;
        asm volatile("global_load_async_to_lds_b128 %0, %1, %2"
                     :: "v"(lds0), "v"(voff), "s"(sa) : "memory");
    }

    for (int c1 = 0; c1 < CC; ++c1) {
        const int buf = c1 & 1;
        if (c1 + 1 < CC) {
            // prefetch next channel into the other buffer, then wait for current
            uint64_t sa  = (uint64_t)(uintptr_t)(srcb + (size_t)(c1 + 1) * (HH * WW));
            uint32_t dst = lds0 + (uint32_t)(buf ^ 1) * (8u * 128u * 4u);
            asm volatile("global_load_async_to_lds_b128 %0, %1, %2"
                         :: "v"(dst), "v"(voff), "s"(sa) : "memory");
            asm volatile("s_wait_asynccnt 1" ::: "memory");
        } else {
            asm volatile("s_wait_asynccnt 0" ::: "memory");
        }
#pragma unroll
        for (int j = 0; j < G; ++j) {
            float v = smem[buf][lidx[j]];                       // ds_load_b32
            __builtin_nontemporal_store(v, orow[j] + (size_t)c1 * PP);
        }
    }
}

extern "C" void kernel_launch(void* const* d_in, const int* in_sizes, int n_in,
                              void* d_out, int out_size, void* d_ws, size_t ws_size,
                              hipStream_t stream) {
    (void)in_sizes; (void)n_in; (void)d_ws; (void)ws_size; (void)out_size;
    const float* x = (const float*)d_in[0];
    float* out = (float*)d_out;
    dim3 grid(PP / PTILE, CC / G, BB);   // (4, 16, 8) = 512 blocks, 8 waves each
    cwmp_gather<<<grid, dim3(256), 0, stream>>>(x, out);
}